// DETRCriterion_81939386073357
// MI455X (gfx1250) — compile-verified
//
#include <hip/hip_runtime.h>
#include <hip/hip_bf16.h>
#include <math.h>

#define B_ 32
#define N_ 300
#define M_ 60
#define L_ 5
#define C_ 1000
#define NB_BCE 40

typedef __attribute__((ext_vector_type(2))) float v2f;
typedef __attribute__((ext_vector_type(8))) float v8f;

// ---------------------------------------------------------------------------
// Wave32 sum reduction via V_WMMA_F32_16X16X4_F32 with an all-ones B matrix.
// A layout (16x4 f32): lanes 0-15 hold A[m][0],A[m][1] (m=lane); lanes 16-31
// hold A[m][2],A[m][3] (m=lane-16). We place each lane's value in its first
// K slot, 0 in the second -> rowsum[m] = x[m] + x[m+16]. With B==ones, every
// D column equals the rowsum, so a lane's 8 C/D registers are 8 distinct rows;
// summing them and xor-shuffling across the half-wave gives the full 32-lane
// sum in full f32 precision. Must be called from converged code (EXEC all 1s).
// ---------------------------------------------------------------------------
__device__ __forceinline__ float wave_reduce_sum_wmma(float x) {
  v2f a; a[0] = x;    a[1] = 0.0f;
  v2f b; b[0] = 1.0f; b[1] = 1.0f;
  v8f c = {};
  c = __builtin_amdgcn_wmma_f32_16x16x4_f32(
      /*neg_a=*/false, a, /*neg_b=*/false, b,
      /*c_mod=*/(short)0, c, /*reuse_a=*/false, /*reuse_b=*/false);
  float s = c[0] + c[1] + c[2] + c[3] + c[4] + c[5] + c[6] + c[7];
  return s + __shfl_xor(s, 16, 32);
}

__device__ __forceinline__ float block_reduce_sum(float x, float* lds8) {
  float w = wave_reduce_sum_wmma(x);
  const int lane = threadIdx.x & 31;
  const int wid  = threadIdx.x >> 5;
  __syncthreads();                      // protect lds8 reuse across calls
  if (lane == 0) lds8[wid] = w;
  __syncthreads();
  const int nw = (int)(blockDim.x >> 5);
  float y = (lane < nw) ? lds8[lane] : 0.0f;
  return wave_reduce_sum_wmma(y);       // all lanes of all waves get the sum
}

__device__ __forceinline__ float block_reduce_max(float x, float* lds8) {
  for (int o = 16; o > 0; o >>= 1) x = fmaxf(x, __shfl_xor(x, o, 32));
  const int lane = threadIdx.x & 31;
  const int wid  = threadIdx.x >> 5;
  __syncthreads();
  if (lane == 0) lds8[wid] = x;
  __syncthreads();
  const int nw = (int)(blockDim.x >> 5);
  float y = (lane < nw) ? lds8[lane] : -INFINITY;
  for (int o = 16; o > 0; o >>= 1) y = fmaxf(y, __shfl_xor(y, o, 32));
  return y;
}

// ---------------------------------------------------------------------------
// Geometry helpers (match reference, f32 like jnp.float32)
// ---------------------------------------------------------------------------
struct Box { float x0, y0, x1, y1; };

__device__ __forceinline__ Box to_xyxy(float cx, float cy, float w, float h) {
  Box b; b.x0 = cx - 0.5f * w; b.y0 = cy - 0.5f * h;
         b.x1 = cx + 0.5f * w; b.y1 = cy + 0.5f * h; return b;
}

__device__ __forceinline__ float giou_f(Box a, Box b) {
  const float eps = 1e-7f;
  float aa = (a.x1 - a.x0) * (a.y1 - a.y0);
  float ab = (b.x1 - b.x0) * (b.y1 - b.y0);
  float ltx = fmaxf(a.x0, b.x0), lty = fmaxf(a.y0, b.y0);
  float rbx = fminf(a.x1, b.x1), rby = fminf(a.y1, b.y1);
  float w = fmaxf(rbx - ltx, 0.0f), h = fmaxf(rby - lty, 0.0f);
  float inter = w * h;
  float uni = aa + ab - inter;
  float iou = inter / (uni + eps);
  float ex0 = fminf(a.x0, b.x0), ey0 = fminf(a.y0, b.y0);
  float ex1 = fmaxf(a.x1, b.x1), ey1 = fmaxf(a.y1, b.y1);
  float ew = fmaxf(ex1 - ex0, 0.0f), eh = fmaxf(ey1 - ey0, 0.0f);
  float enc = ew * eh;
  return iou - (enc - uni) / (enc + eps);
}

__device__ __forceinline__ float bce_f(float x, float t) {
  return fmaxf(x, 0.0f) - x * t + log1pf(expf(-fabsf(x)));
}

// ---------------------------------------------------------------------------
// Kernel 1: Hungarian (Jonker-Volgenant) per batch element, entirely in LDS.
// Cost matrix (n_valid_targets x 300 preds, i.e. the reference's transposed
// orientation) lives in LDS (<=72KB). Potentials in f64 to match numpy.
// Emits match lists (pidx/tidx/k) and the dense objectness target map.
// ---------------------------------------------------------------------------
__global__ void __launch_bounds__(256) k_hungarian(
    const float* __restrict__ pred, const float* __restrict__ tgt,
    const float* __restrict__ mask,
    int* __restrict__ pidx, int* __restrict__ tidx, int* __restrict__ kcnt,
    float* __restrict__ obj_t)
{
  extern __shared__ char smem_raw[];
  double* u    = (double*)smem_raw;          // 64
  double* v    = u + 64;                     // 304
  double* minv = v + 304;                    // 304
  double* rvd  = minv + 304;                 // 256 (argmin reduce values)
  float*  cost = (float*)(rvd + 256);        // 60*300 = 18000
  int* p    = (int*)(cost + 18000);          // 304
  int* way  = p + 304;                       // 304
  int* used = way + 304;                     // 304
  int* vid  = used + 304;                    // 64
  int* ridx = vid + 64;                      // 256 (argmin reduce indices)
  int* misc = ridx + 256;                    // 8

  const int b   = blockIdx.x;
  const int tid = threadIdx.x;
  const int T   = blockDim.x;

  // valid target column indices (order preserved, matches np.nonzero)
  if (tid == 0) {
    int n = 0;
    for (int m = 0; m < M_; ++m)
      if (mask[b * M_ + m] > 0.5f) vid[n++] = m;
    misc[1] = n;
  }
  __syncthreads();
  const int n = misc[1];

  // cost[r][c] = 5*L1 + 2*(-giou), r = valid-target row, c = pred column
  for (int e = tid; e < n * N_; e += T) {
    const int r = e / N_, c = e % N_;
    const float* pp = pred + ((size_t)b * N_ + c) * 4;
    const float* tt = tgt  + ((size_t)b * M_ + vid[r]) * 4;
    float l1 = fabsf(pp[0] - tt[0]) + fabsf(pp[1] - tt[1]) +
               fabsf(pp[2] - tt[2]) + fabsf(pp[3] - tt[3]);
    float g = giou_f(to_xyxy(pp[0], pp[1], pp[2], pp[3]),
                     to_xyxy(tt[0], tt[1], tt[2], tt[3]));
    cost[r * N_ + c] = 5.0f * l1 - 2.0f * g;
  }
  for (int j = tid; j <= N_; j += T) { v[j] = 0.0; p[j] = 0; way[j] = 0; }
  for (int j = tid; j <= M_; j += T) u[j] = 0.0;
  __syncthreads();

  for (int i = 1; i <= n; ++i) {
    if (tid == 0) { p[0] = i; misc[0] = 0; }
    for (int j = tid; j <= N_; j += T) { minv[j] = INFINITY; used[j] = 0; }
    __syncthreads();

    for (;;) {
      const int j0 = misc[0];
      if (tid == 0) used[j0] = 1;
      __syncthreads();
      const int    i0  = p[j0];
      const double ui0 = u[i0];

      // relax free columns (parallel over 300 cols)
      for (int j = 1 + tid; j <= N_; j += T) {
        if (!used[j]) {
          double cur = (double)cost[(i0 - 1) * N_ + (j - 1)] - ui0 - v[j];
          if (cur < minv[j]) { minv[j] = cur; way[j] = j0; }
        }
      }
      __syncthreads();

      // argmin over free columns; tie -> smallest index (np.argmin semantics)
      double mv = INFINITY; int mj = 0x7fffffff;
      for (int j = 1 + tid; j <= N_; j += T)
        if (!used[j]) { double m = minv[j]; if (m < mv) { mv = m; mj = j; } }
      rvd[tid] = mv; ridx[tid] = mj;
      __syncthreads();
      for (int s = T >> 1; s > 0; s >>= 1) {
        if (tid < s) {
          double ov = rvd[tid + s]; int oi = ridx[tid + s];
          if (ov < rvd[tid] || (ov == rvd[tid] && oi < ridx[tid])) {
            rvd[tid] = ov; ridx[tid] = oi;
          }
        }
        __syncthreads();
      }
      const double delta = rvd[0];
      const int    j1    = ridx[0];
      __syncthreads();

      // dual updates (p[j] distinct over used columns -> race-free)
      for (int j = tid; j <= N_; j += T) {
        if (used[j]) { u[p[j]] += delta; v[j] -= delta; }
        else         { minv[j] -= delta; }
      }
      if (tid == 0) misc[0] = j1;
      __syncthreads();
      if (p[j1] == 0) break;
    }

    if (tid == 0) {  // augment along the alternating path
      int j0 = misc[0];
      while (j0) { int jn = way[j0]; p[j0] = p[jn]; j0 = jn; }
    }
    __syncthreads();
  }

  // emit matches + dense objectness targets (deterministic order)
  for (int j = tid; j < N_; j += T) obj_t[b * N_ + j] = 0.0f;
  __syncthreads();
  if (tid == 0) {
    int k = 0;
    for (int j = 1; j <= N_; ++j) {
      const int r = p[j];
      if (r > 0) {
        pidx[b * M_ + k] = j - 1;
        tidx[b * M_ + k] = vid[r - 1];
        obj_t[b * N_ + (j - 1)] = 1.0f;
        ++k;
      }
    }
    kcnt[b] = k;
  }
}

// ---------------------------------------------------------------------------
// Kernel 2: per-batch -log_softmax picked at label (logsumexp - logit[label])
// ---------------------------------------------------------------------------
__global__ void __launch_bounds__(256) k_cls(
    const float* __restrict__ logits, const int* __restrict__ label,
    float* __restrict__ cls_p)
{
  __shared__ float lds8[8];
  const int b = blockIdx.x, tid = threadIdx.x;
  const float* row = logits + (size_t)b * C_;
  float mx = -INFINITY;
  for (int i = tid; i < C_; i += blockDim.x) mx = fmaxf(mx, row[i]);
  mx = block_reduce_max(mx, lds8);
  float se = 0.0f;
  for (int i = tid; i < C_; i += blockDim.x) se += expf(row[i] - mx);
  se = block_reduce_sum(se, lds8);
  if (tid == 0) cls_p[b] = (mx + logf(se)) - row[label[b]];
}

// ---------------------------------------------------------------------------
// Kernel 3: matched-pair L1 and (1 - GIoU) sums, main head + aux heads
// ---------------------------------------------------------------------------
__global__ void __launch_bounds__(256) k_matched(
    const float* __restrict__ pred, const float* __restrict__ tgt,
    const float* __restrict__ auxp,
    const int* __restrict__ pidx, const int* __restrict__ tidx,
    const int* __restrict__ kcnt,
    float* __restrict__ mb_p, float* __restrict__ ab_p)
{
  __shared__ float lds8[8];
  const int b = blockIdx.x, tid = threadIdx.x;
  const int k = kcnt[b];
  float bb = 0.0f, gg = 0.0f, abb = 0.0f, agg = 0.0f;
  if (tid < k) {
    const int pj = pidx[b * M_ + tid];
    const int tj = tidx[b * M_ + tid];
    const float* pp = pred + ((size_t)b * N_ + pj) * 4;
    const float* tt = tgt  + ((size_t)b * M_ + tj) * 4;
    const Box tb = to_xyxy(tt[0], tt[1], tt[2], tt[3]);
    bb = fabsf(pp[0] - tt[0]) + fabsf(pp[1] - tt[1]) +
         fabsf(pp[2] - tt[2]) + fabsf(pp[3] - tt[3]);
    gg = 1.0f - giou_f(to_xyxy(pp[0], pp[1], pp[2], pp[3]), tb);
    for (int l = 0; l < L_; ++l) {
      const float* ap = auxp + (((size_t)l * B_ + b) * N_ + pj) * 4;
      abb += fabsf(ap[0] - tt[0]) + fabsf(ap[1] - tt[1]) +
             fabsf(ap[2] - tt[2]) + fabsf(ap[3] - tt[3]);
      agg += 1.0f - giou_f(to_xyxy(ap[0], ap[1], ap[2], ap[3]), tb);
    }
  }
  bb  = block_reduce_sum(bb,  lds8);
  gg  = block_reduce_sum(gg,  lds8);
  abb = block_reduce_sum(abb, lds8);
  agg = block_reduce_sum(agg, lds8);
  if (tid == 0) {
    mb_p[2 * b] = bb;  mb_p[2 * b + 1] = gg;
    ab_p[2 * b] = abb; ab_p[2 * b + 1] = agg;
  }
}

// ---------------------------------------------------------------------------
// Kernel 4: BCE sums for main objectness (B*N) and aux objectness (L*B*N)
// ---------------------------------------------------------------------------
__global__ void __launch_bounds__(256) k_bce(
    const float* __restrict__ obj, const float* __restrict__ auxo,
    const float* __restrict__ obj_t, float* __restrict__ bce_p)
{
  __shared__ float lds8[8];
  const int tid = threadIdx.x;
  float am = 0.0f, aa = 0.0f;
  const int bn = B_ * N_;
  const int total = bn * (1 + L_);
  for (int idx = blockIdx.x * blockDim.x + tid; idx < total;
       idx += gridDim.x * blockDim.x) {
    if (idx < bn) {
      am += bce_f(obj[idx], obj_t[idx]);
    } else {
      const int e = idx - bn;
      aa += bce_f(auxo[e], obj_t[e % bn]);
    }
  }
  am = block_reduce_sum(am, lds8);
  aa = block_reduce_sum(aa, lds8);
  if (tid == 0) { bce_p[2 * blockIdx.x] = am; bce_p[2 * blockIdx.x + 1] = aa; }
}

// ---------------------------------------------------------------------------
// Kernel 5: combine all partials into the scalar loss
// ---------------------------------------------------------------------------
__global__ void __launch_bounds__(256) k_final(
    const float* __restrict__ cls_p, const int* __restrict__ kcnt,
    const float* __restrict__ mb_p, const float* __restrict__ ab_p,
    const float* __restrict__ bce_p, float* __restrict__ out)
{
  __shared__ float lds8[8];
  const int tid = threadIdx.x;
  float cs  = block_reduce_sum((tid < B_)     ? cls_p[tid]            : 0.0f, lds8);
  float ks  = block_reduce_sum((tid < B_)     ? (float)kcnt[tid]      : 0.0f, lds8);
  float mbb = block_reduce_sum((tid < B_)     ? mb_p[2 * tid]         : 0.0f, lds8);
  float mgi = block_reduce_sum((tid < B_)     ? mb_p[2 * tid + 1]     : 0.0f, lds8);
  float abb = block_reduce_sum((tid < B_)     ? ab_p[2 * tid]         : 0.0f, lds8);
  float agi = block_reduce_sum((tid < B_)     ? ab_p[2 * tid + 1]     : 0.0f, lds8);
  float bm  = block_reduce_sum((tid < NB_BCE) ? bce_p[2 * tid]        : 0.0f, lds8);
  float ba  = block_reduce_sum((tid < NB_BCE) ? bce_p[2 * tid + 1]    : 0.0f, lds8);
  if (tid == 0) {
    const float nb     = fmaxf(ks, 1.0f);
    const float inv_bn = 1.0f / (float)(B_ * N_);
    const float main_l = 5.0f * (mbb / nb) + 2.0f * (mgi / nb) + 1.0f * (bm * inv_bn);
    const float aux_l  = (5.0f * (abb / nb) + 2.0f * (agi / nb) + 1.0f * (ba * inv_bn))
                         * (0.5f / (float)L_);
    out[0] = cs / (float)B_ + main_l + aux_l;
  }
}

// ---------------------------------------------------------------------------
extern "C" void kernel_launch(void* const* d_in, const int* in_sizes, int n_in,
                              void* d_out, int out_size, void* d_ws, size_t ws_size,
                              hipStream_t stream) {
  (void)in_sizes; (void)n_in; (void)out_size; (void)ws_size;

  const float* cls_logits = (const float*)d_in[0];
  const int*   label      = (const int*)  d_in[1];
  const float* pred       = (const float*)d_in[2];
  const float* obj        = (const float*)d_in[3];
  const float* tgt        = (const float*)d_in[4];
  const float* mask       = (const float*)d_in[5];
  const float* auxp       = (const float*)d_in[6];
  const float* auxo       = (const float*)d_in[7];
  float* out = (float*)d_out;

  // workspace layout (~55KB): match lists, obj targets, partial sums
  char* ws = (char*)d_ws;
  int*   pidx  = (int*)ws;               // B*M
  int*   tidx  = pidx + B_ * M_;         // B*M
  int*   kcnt  = tidx + B_ * M_;         // B
  float* obj_t = (float*)(kcnt + B_);    // B*N
  float* cls_p = obj_t + B_ * N_;        // B
  float* mb_p  = cls_p + B_;             // 2*B
  float* ab_p  = mb_p + 2 * B_;          // 2*B
  float* bce_p = ab_p + 2 * B_;          // 2*NB_BCE

  // dynamic LDS for Hungarian: f64 potentials + f32 cost tile + int state
  const size_t sh = (size_t)(64 + 304 + 304 + 256) * sizeof(double)
                  + (size_t)18000 * sizeof(float)
                  + (size_t)(304 * 3 + 64 + 256 + 8) * sizeof(int);

  k_hungarian<<<B_, 256, sh, stream>>>(pred, tgt, mask, pidx, tidx, kcnt, obj_t);
  k_cls     <<<B_, 256, 0, stream>>>(cls_logits, label, cls_p);
  k_matched <<<B_, 256, 0, stream>>>(pred, tgt, auxp, pidx, tidx, kcnt, mb_p, ab_p);
  k_bce     <<<NB_BCE, 256, 0, stream>>>(obj, auxo, obj_t, bce_p);
  k_final   <<<1, 256, 0, stream>>>(cls_p, kcnt, mb_p, ab_p, bce_p, out);
}